// DWT_2D_FFT_34763465294182
// MI455X (gfx1250) — compile-verified
//
#include <hip/hip_runtime.h>

// ---------------------------------------------------------------------------
// 2D db2 DWT (circular, stride-2) for x:(8,96,128,128) f32.
// out = [LL | LH | HL | HH], each (8,96,64,64), concatenated flat.
// One workgroup per (b,c) image. Async-to-LDS load (ASYNCcnt path), row conv
// in VALU, column conv + subsample via v_wmma_f32_16x16x4_f32 banded matmuls.
// Memory-bound: ~100 MB total traffic -> ~4.3 us floor at 23.3 TB/s.
// ---------------------------------------------------------------------------

typedef float v2f __attribute__((ext_vector_type(2)));
typedef float v8f __attribute__((ext_vector_type(8)));
typedef int   v4i __attribute__((vector_size(16)));   // matches builtin proto

#define IMG_SZ   16384          // 128*128
#define N_IMG    768            // 8*96
#define OUT_HW   4096           // 64*64
#define QSTRIDE  (N_IMG * OUT_HW)
#define XL_PITCH 65             // padded row pitch (floats) to avoid LDS bank conflicts

__device__ __forceinline__ void async_ld16(const float* g, float* l) {
#if __has_builtin(__builtin_amdgcn_global_load_async_to_lds_b128)
  __builtin_amdgcn_global_load_async_to_lds_b128(
      (__attribute__((address_space(1))) v4i*)(void*)g,
      (__attribute__((address_space(3))) v4i*)(void*)l,
      0, 0);
#else
  *(float4*)l = *(const float4*)g;   // fallback: direct 16B copy via VGPRs
#endif
}

__device__ __forceinline__ void wait_async() {
#if __has_builtin(__builtin_amdgcn_s_wait_asynccnt)
  __builtin_amdgcn_s_wait_asynccnt(0);
#elif __has_builtin(__builtin_amdgcn_global_load_async_to_lds_b128)
  asm volatile("s_wait_asynccnt 0" ::: "memory");
#endif
}

// Select tap j from {f0..f3}, 0 outside the band (VALU cndmask chain).
__device__ __forceinline__ float tap(float f0, float f1, float f2, float f3, int j) {
  float r = 0.0f;
  r = (j == 0) ? f0 : r;
  r = (j == 1) ? f1 : r;
  r = (j == 2) ? f2 : r;
  r = (j == 3) ? f3 : r;
  return r;
}

__global__ __launch_bounds__(256)
void dwt2d_db2_kernel(const float* __restrict__ x,
                      const float* __restrict__ wl,
                      const float* __restrict__ wh,
                      float* __restrict__ out) {
  extern __shared__ float smem[];
  float* sX  = smem;                         // [128][128] raw image
  float* sXL = smem + IMG_SZ;                // [128][XL_PITCH] row-lowpass @ even cols
  float* sXH = sXL + 128 * XL_PITCH;         // [128][XL_PITCH] row-highpass @ even cols

  const int t   = threadIdx.x;
  const int img = blockIdx.x;
  const float* src = x + (size_t)img * IMG_SZ;

  // Effective taps: out[n] = F0*x[n-1] + F1*x[n] + F2*x[n+1] + F3*x[n+2],
  // with F[j] = w[3-j] (FFT-multiply + roll(-L/2) of the reversed filter).
  const float l0 = wl[3], l1 = wl[2], l2 = wl[1], l3 = wl[0];
  const float h0 = wh[3], h1 = wh[2], h2 = wh[1], h3 = wh[0];

  // ---- Phase 1: stream the 64 KB image into LDS (async, ASYNCcnt-tracked) ----
#pragma unroll
  for (int i = 0; i < 16; ++i) {
    const int chunk = i * 256 + t;           // 16-byte chunks, 4096 total
    async_ld16(src + chunk * 4, sX + chunk * 4);
  }
  wait_async();
  __syncthreads();

  // ---- Phase 2: row (W) circular conv, keep even columns only ----
  for (int o = t; o < 8192; o += 256) {      // 128 rows x 64 even cols
    const int h = o >> 6, w2 = o & 63, w = w2 * 2;
    const float* row = sX + h * 128;
    const float a = row[(w - 1) & 127];
    const float b = row[w];
    const float c = row[w + 1];
    const float d = row[(w + 2) & 127];
    sXL[h * XL_PITCH + w2] = fmaf(l0, a, fmaf(l1, b, fmaf(l2, c, l3 * d)));
    sXH[h * XL_PITCH + w2] = fmaf(h0, a, fmaf(h1, b, fmaf(h2, c, h3 * d)));
  }
  __syncthreads();

  // ---- Phase 3: column (H) conv + subsample via WMMA f32 16x16x4 ----
  // Output tile D(16x16) = sum_{c=0..8} A_c(16x4, banded filter) x B_c(4x16).
  // A layout (ISA 7.12.2): lane<16 holds {K=0,K=1} of row M=lane,
  //                        lane>=16 holds {K=2,K=3} of row M=lane-16.
  // B layout mirrors A with M->N.
  const int lane = t & 31;
  const int wv   = t >> 5;
  const int ip   = lane & 15;                // A row / B column
  const int kb   = (lane >> 4) * 2;          // K base for this lane half

  for (int task = wv; task < 32; task += 8) {
    const int   srcSel = task >> 4;          // 0: xL -> LL,LH   1: xH -> HL,HH
    const float* S     = srcSel ? sXH : sXL;
    const int   pos = task & 15;
    const int   i0  = (pos >> 2) * 16;       // output row block
    const int   w0  = (pos & 3) * 16;        // output col block
    const int   rbase = 2 * i0 - 1;          // first needed input row (mod 128)

    v8f accA = {};                           // low-pass column filter
    v8f accB = {};                           // high-pass column filter

#pragma unroll
    for (int c = 0; c < 9; ++c) {
      const int m0 = 4 * c + kb;
      const int j0 = m0 - 2 * ip;            // band offset: A[i',m] = F[m - 2i']
      v2f Alo, Ahi, Bv;
      Alo.x = tap(l0, l1, l2, l3, j0);
      Alo.y = tap(l0, l1, l2, l3, j0 + 1);
      Ahi.x = tap(h0, h1, h2, h3, j0);
      Ahi.y = tap(h0, h1, h2, h3, j0 + 1);
      const int r0 = (rbase + m0)     & 127;
      const int r1 = (rbase + m0 + 1) & 127;
      Bv.x = S[r0 * XL_PITCH + w0 + ip];
      Bv.y = S[r1 * XL_PITCH + w0 + ip];
      accA = __builtin_amdgcn_wmma_f32_16x16x4_f32(
          false, Alo, false, Bv, (short)0, accA, false, false);
      accB = __builtin_amdgcn_wmma_f32_16x16x4_f32(
          false, Ahi, false, Bv, (short)0, accB, false, false);
    }

    // C/D layout: VGPR v -> M = v (lanes 0-15) or M = v+8 (lanes 16-31), N = lane&15.
    float* oA = out + (size_t)(srcSel * 2) * QSTRIDE + (size_t)img * OUT_HW;
    float* oB = oA + QSTRIDE;
    const int rofs = i0 + (lane >> 4) * 8;
    const int col  = w0 + ip;
#pragma unroll
    for (int v = 0; v < 8; ++v) {
      oA[(rofs + v) * 64 + col] = accA[v];
      oB[(rofs + v) * 64 + col] = accB[v];
    }
  }
}

extern "C" void kernel_launch(void* const* d_in, const int* in_sizes, int n_in,
                              void* d_out, int out_size, void* d_ws, size_t ws_size,
                              hipStream_t stream) {
  const float* x  = (const float*)d_in[0];
  const float* wl = (const float*)d_in[1];
  const float* wh = (const float*)d_in[2];
  float* out = (float*)d_out;

  const size_t shmem = (IMG_SZ + 2 * 128 * XL_PITCH) * sizeof(float); // ~129 KB
  dwt2d_db2_kernel<<<dim3(N_IMG), dim3(256), shmem, stream>>>(x, wl, wh, out);
}